// GCNEncoder_5162550690708
// MI455X (gfx1250) — compile-verified
//
#include <hip/hip_runtime.h>

// GCN 2-layer: per-layer  out = scatter(norm * (X@W)) + selfloop + bias (+relu)
// MI455X: fp32 WMMA 16x16x4 for the GEMMs, LDS-staged W, one-wave-per-edge
// scatter with f32 atomics (L2-resident accumulators, 192MB L2).

typedef float v2f __attribute__((ext_vector_type(2)));
typedef float v8f __attribute__((ext_vector_type(8)));

#define IN_C 128
#define HID_C 64

// ---------------- degree / norm ----------------
__global__ void deg_kernel(const long long* __restrict__ dst,
                           unsigned* __restrict__ deg, int E) {
    int e = blockIdx.x * blockDim.x + threadIdx.x;
    if (e < E) atomicAdd(&deg[(int)dst[e]], 1u);
}

__global__ void dinv_kernel(const unsigned* __restrict__ deg,
                            float* __restrict__ dinv, int N) {
    int i = blockIdx.x * blockDim.x + threadIdx.x;
    if (i < N) dinv[i] = rsqrtf((float)(deg[i] + 1u));   // +1 self-loop
}

// ---------------- fp32 WMMA GEMM:  Out[N,64] = X[N,K] @ W[K,64] ----------------
// 256 threads = 8 waves. Wave w -> Mtile = blk*2 + w/4, Ntile = w%4.
// W staged in LDS (K*64*4 <= 32KB).
template <int K>
__global__ __launch_bounds__(256) void gemm_wmma(const float* __restrict__ X,
                                                 const float* __restrict__ W,
                                                 float* __restrict__ Out,
                                                 int Nnodes) {
    __shared__ float Wlds[K * 64];
    const int tid = threadIdx.x;
    for (int i = tid; i < K * 64; i += 256) Wlds[i] = W[i];
    __syncthreads();

    const int wave  = tid >> 5;
    const int lane  = tid & 31;
    const int mtile = blockIdx.x * 2 + (wave >> 2);
    const int ntile = wave & 3;
    if (mtile * 16 >= Nnodes) return;

    const int m16  = lane & 15;            // M (A) or N (B/C/D) within tile
    const int half = lane >> 4;            // lane group 0/1
    const int koff = half * 2;             // A/B VGPR K offset per ISA layout
    const int col  = ntile * 16 + m16;

    const float* __restrict__ xrow = X + (long)(mtile * 16 + m16) * K + koff;

    v8f acc = {};
    #pragma unroll 4
    for (int k = 0; k < K; k += 4) {
        // A 16x4 fp32: lane m16 holds rows M=m16, VGPR0/1 = K = koff, koff+1
        v2f a = *(const v2f*)(xrow + k);
        // B 4x16 fp32 (mirrored layout): lane holds col, VGPR0/1 = K = koff, koff+1
        v2f b;
        b.x = Wlds[(k + koff) * 64 + col];
        b.y = Wlds[(k + koff + 1) * 64 + col];
        acc = __builtin_amdgcn_wmma_f32_16x16x4_f32(
            /*neg_a=*/false, a, /*neg_b=*/false, b,
            /*c_mod=*/(short)0, acc, /*reuse_a=*/false, /*reuse_b=*/false);
    }

    // C/D 16x16 fp32: VGPR i -> row M = i + 8*half, lanes give N = col
    float* __restrict__ orow = Out + (long)(mtile * 16 + half * 8) * 64 + col;
    #pragma unroll
    for (int i = 0; i < 8; ++i) orow[(long)i * 64] = acc[i];
}

// ---------------- edge scatter: one wave per edge, 2 channels per lane ----------------
__global__ void scatter_kernel(const long long* __restrict__ src,
                               const long long* __restrict__ dst,
                               const float* __restrict__ dinv,
                               const float* __restrict__ H,
                               float* __restrict__ Out, int E) {
    long long gid = (long long)blockIdx.x * blockDim.x + threadIdx.x;
    int e = (int)(gid >> 5);
    if (e >= E) return;
    int c2 = ((int)gid & 31) * 2;
    int s = (int)src[e];
    int d = (int)dst[e];
    float norm = dinv[s] * dinv[d];
    v2f v = *(const v2f*)(H + (long)s * 64 + c2);
    atomicAdd(Out + (long)d * 64 + c2,     v.x * norm);
    atomicAdd(Out + (long)d * 64 + c2 + 1, v.y * norm);
}

// ---------------- finalize: self-loop + bias (+relu) ----------------
__global__ void finalize_kernel(float* __restrict__ Agg,
                                const float* __restrict__ XW,
                                const float* __restrict__ dinv,
                                const float* __restrict__ bias,
                                int N, int do_relu) {
    int idx = blockIdx.x * blockDim.x + threadIdx.x;
    if (idx >= N * 64) return;
    int node = idx >> 6;
    int c    = idx & 63;
    float di = dinv[node];
    float v  = Agg[idx] + XW[idx] * di * di + bias[c];
    Agg[idx] = do_relu ? fmaxf(v, 0.0f) : v;
}

extern "C" void kernel_launch(void* const* d_in, const int* in_sizes, int n_in,
                              void* d_out, int out_size, void* d_ws, size_t ws_size,
                              hipStream_t stream) {
    (void)n_in; (void)out_size; (void)ws_size;

    const float*     x   = (const float*)d_in[0];
    const long long* ei  = (const long long*)d_in[1];   // int64 [2, E]
    const float*     W1  = (const float*)d_in[2];
    const float*     b1  = (const float*)d_in[3];
    const float*     W2  = (const float*)d_in[4];
    const float*     b2  = (const float*)d_in[5];

    const int N = in_sizes[0] / IN_C;      // 100000
    const int E = in_sizes[1] / 2;         // 1600000
    const long long* src = ei;
    const long long* dst = ei + E;

    // workspace layout
    char*     ws   = (char*)d_ws;
    unsigned* deg  = (unsigned*)ws;                                     // N u32
    float*    dinv = (float*)(ws + (size_t)N * 4);                      // N f32
    float*    XW   = (float*)(ws + (size_t)N * 8);                      // N*64 f32
    float*    AGG  = (float*)(ws + (size_t)N * 8 + (size_t)N * 64 * 4); // N*64 f32
    float*    out  = (float*)d_out;

    // zero accumulators every call (deterministic)
    hipMemsetAsync(deg, 0, (size_t)N * 4, stream);
    hipMemsetAsync(AGG, 0, (size_t)N * 64 * 4, stream);
    hipMemsetAsync(out, 0, (size_t)N * 64 * 4, stream);

    const int B = 256;
    deg_kernel<<<(E + B - 1) / B, B, 0, stream>>>(dst, deg, E);
    dinv_kernel<<<(N + B - 1) / B, B, 0, stream>>>(deg, dinv, N);

    // layer 1
    gemm_wmma<IN_C><<<(N + 31) / 32, B, 0, stream>>>(x, W1, XW, N);
    {
        long long t = (long long)E * 32;
        scatter_kernel<<<(int)((t + B - 1) / B), B, 0, stream>>>(src, dst, dinv, XW, AGG, E);
    }
    finalize_kernel<<<(N * 64 + B - 1) / B, B, 0, stream>>>(AGG, XW, dinv, b1, N, 1);

    // layer 2
    gemm_wmma<HID_C><<<(N + 31) / 32, B, 0, stream>>>(AGG, W2, XW, N);
    {
        long long t = (long long)E * 32;
        scatter_kernel<<<(int)((t + B - 1) / B), B, 0, stream>>>(src, dst, dinv, XW, out, E);
    }
    finalize_kernel<<<(N * 64 + B - 1) / B, B, 0, stream>>>(out, XW, dinv, b2, N, 0);
}